// RGCNLayer_68049461838045
// MI455X (gfx1250) — compile-verified
//
#include <hip/hip_runtime.h>
#include <hip/hip_bf16.h>

typedef __attribute__((ext_vector_type(2))) float v2f;
typedef __attribute__((ext_vector_type(8))) float v8f;

#define IN_DIM   64
#define OUT_DIM  64
#define NUM_RELS 64
#define NUM_BASES 8
#define WAVES    4   // waves per block (128 threads, wave32)

// ---------------------------------------------------------------------------
// K0: zero degree + relation histogram
__global__ void rgcn_zero_kernel(float* deg, int* counts, int N) {
    int i = blockIdx.x * blockDim.x + threadIdx.x;
    if (i < N) deg[i] = 0.0f;
    if (i < NUM_RELS) counts[i] = 0;
}

// K1: in-degree (float) + relation histogram
__global__ void rgcn_deg_hist_kernel(const int* dst, const int* etype,
                                     float* deg, int* counts, int E) {
    int e = blockIdx.x * blockDim.x + threadIdx.x;
    if (e < E) {
        atomicAdd(&deg[dst[e]], 1.0f);
        atomicAdd(&counts[etype[e]], 1);
    }
}

// K2: norm = 1/(deg+1), in place
__global__ void rgcn_norm_kernel(float* deg, int N) {
    int i = blockIdx.x * blockDim.x + threadIdx.x;
    if (i < N) deg[i] = 1.0f / (deg[i] + 1.0f);
}

// K3: serial scan over 64 bins (trivial)
__global__ void rgcn_scan_kernel(const int* counts, int* bin_start,
                                 int* tile_off, int* cursor) {
    if (blockIdx.x == 0 && threadIdx.x == 0) {
        int bs = 0, ts = 0;
        for (int r = 0; r < NUM_RELS; ++r) {
            bin_start[r] = bs;
            cursor[r]    = bs;
            tile_off[r]  = ts;
            bs += counts[r];
            ts += (counts[r] + 15) >> 4;
        }
        bin_start[NUM_RELS] = bs;
        tile_off[NUM_RELS]  = ts;
    }
}

// K4: scatter edge ids into relation-sorted order
__global__ void rgcn_scatter_kernel(const int* etype, int* cursor,
                                    int* sorted, int E) {
    int e = blockIdx.x * blockDim.x + threadIdx.x;
    if (e < E) {
        int r = etype[e];
        int pos = atomicAdd(&cursor[r], 1);
        sorted[pos] = e;
    }
}

// K5: build per-relation weights W_r = sum_b w_comp[r,b]*B_b, packed directly
// into the WMMA B-fragment layout:
//   packed[((k*4 + ntile)*32 + lane)*2 + v] = W[4k + 2*(lane>>4) + v][ntile*16 + (lane&15)]
// blockIdx.x==NUM_RELS packs loop_weight the same way.
__global__ void rgcn_pack_kernel(const float* basis, const float* w_comp,
                                 const float* loopw, float* relpack,
                                 float* looppack) {
    int r = blockIdx.x;
    for (int t = threadIdx.x; t < 4096; t += blockDim.x) {
        int k    = t >> 8;
        int nt   = (t >> 6) & 3;
        int lane = (t >> 1) & 31;
        int v    = t & 1;
        int i = (k << 2) + ((lane >> 4) << 1) + v;   // K (input dim)
        int o = (nt << 4) + (lane & 15);             // N (output dim)
        if (r < NUM_RELS) {
            float s = 0.0f;
            #pragma unroll
            for (int b = 0; b < NUM_BASES; ++b)
                s += w_comp[r * NUM_BASES + b] * basis[((size_t)b * IN_DIM + i) * OUT_DIM + o];
            relpack[(size_t)r * 4096 + t] = s;
        } else {
            looppack[t] = loopw[(size_t)i * OUT_DIM + o];
        }
    }
}

// K6: self-loop GEMM via WMMA: agg = (node_feat @ loop_weight) * norm + bias
__global__ void rgcn_selfloop_kernel(const float* __restrict__ node_feat,
                                     const float* __restrict__ norm,
                                     const float* __restrict__ looppack,
                                     const float* __restrict__ bias,
                                     float* __restrict__ agg, int N) {
    __shared__ float Alds[WAVES][16 * IN_DIM];
    __shared__ float normlds[WAVES][16];

    int lane = threadIdx.x & 31;
    int w    = threadIdx.x >> 5;
    int tile = blockIdx.x * WAVES + w;
    int nTiles = (N + 15) >> 4;
    bool active = tile < nTiles;          // wave-uniform
    int base = tile << 4;
    int cnt  = active ? min(16, N - base) : 0;

    if (active && lane < 16)
        normlds[w][lane] = (lane < cnt) ? norm[base + lane] : 0.0f;

    if (active) {
        int m    = lane >> 1;
        int colb = (lane & 1) * 32;
        bool rv  = (m < cnt);
        const float4* gp = reinterpret_cast<const float4*>(
            node_feat + (size_t)(base + (rv ? m : 0)) * IN_DIM + colb);
        float4* lp = reinterpret_cast<float4*>(&Alds[w][m * IN_DIM + colb]);
        #pragma unroll
        for (int j = 0; j < 8; ++j) {
            float4 val = rv ? gp[j] : make_float4(0.f, 0.f, 0.f, 0.f);
            lp[j] = val;
        }
    }
    __syncthreads();

    v8f acc[4] = {};
    int mA  = lane & 15;
    int kHi = (lane >> 4) << 1;
    #pragma unroll 4
    for (int k = 0; k < 16; ++k) {
        v2f a = *reinterpret_cast<const v2f*>(&Alds[w][mA * IN_DIM + (k << 2) + kHi]);
        #pragma unroll
        for (int nt = 0; nt < 4; ++nt) {
            v2f b = *reinterpret_cast<const v2f*>(
                looppack + ((((k << 2) + nt) << 5) + lane) * 2);
            acc[nt] = __builtin_amdgcn_wmma_f32_16x16x4_f32(
                false, a, false, b, (short)0, acc[nt], false, false);
        }
    }

    if (active) {
        #pragma unroll
        for (int nt = 0; nt < 4; ++nt) {
            int n = (nt << 4) + (lane & 15);
            float bn = bias[n];
            #pragma unroll
            for (int v = 0; v < 8; ++v) {
                int m = v + ((lane >> 4) << 3);
                if (m < cnt)
                    agg[(size_t)(base + m) * OUT_DIM + n] =
                        acc[nt][v] * normlds[w][m] + bn;
            }
        }
    }
}

// K7: edge messages via WMMA on relation-sorted 16-edge tiles, atomic scatter
__global__ void rgcn_edge_kernel(const float* __restrict__ node_feat,
                                 const int* __restrict__ src,
                                 const int* __restrict__ dst,
                                 const float* __restrict__ norm,
                                 const int* __restrict__ sorted,
                                 const int* __restrict__ bin_start,
                                 const int* __restrict__ tile_off,
                                 const float* __restrict__ relpack,
                                 float* __restrict__ agg) {
    __shared__ float Alds[WAVES][16 * IN_DIM];
    __shared__ float normlds[WAVES][16];
    __shared__ int   dstlds[WAVES][16];
    __shared__ int   srclds[WAVES][16];

    int lane = threadIdx.x & 31;
    int w    = threadIdx.x >> 5;
    int tile = blockIdx.x * WAVES + w;
    int nTiles = tile_off[NUM_RELS];
    bool active = tile < nTiles;          // wave-uniform

    int r = 0, estart = 0, cnt = 0;
    if (active) {
        while (tile >= tile_off[r + 1]) ++r;   // 64 bins, cheap uniform scan
        int tin = tile - tile_off[r];
        estart  = bin_start[r] + (tin << 4);
        cnt     = bin_start[r + 1] - estart;
        if (cnt > 16) cnt = 16;
    }

    if (active && lane < 16) {
        int m = lane;
        if (m < cnt) {
            int e = sorted[estart + m];
            int s = src[e];
            int d = dst[e];
            srclds[w][m]  = s;
            dstlds[w][m]  = d;
            normlds[w][m] = norm[d];
        } else {
            srclds[w][m]  = 0;
            dstlds[w][m]  = 0;
            normlds[w][m] = 0.0f;
        }
    }
    __syncthreads();

    if (active) {
        int m    = lane >> 1;
        int colb = (lane & 1) * 32;
        bool rv  = (m < cnt);
        const float4* gp = reinterpret_cast<const float4*>(
            node_feat + (size_t)srclds[w][m] * IN_DIM + colb);
        float4* lp = reinterpret_cast<float4*>(&Alds[w][m * IN_DIM + colb]);
        #pragma unroll
        for (int j = 0; j < 8; ++j) {
            float4 val = rv ? gp[j] : make_float4(0.f, 0.f, 0.f, 0.f);
            lp[j] = val;
        }
    }
    __syncthreads();

    v8f acc[4] = {};
    int mA  = lane & 15;
    int kHi = (lane >> 4) << 1;
    const float* bbase = relpack + (size_t)r * 4096;
    #pragma unroll 4
    for (int k = 0; k < 16; ++k) {
        v2f a = *reinterpret_cast<const v2f*>(&Alds[w][mA * IN_DIM + (k << 2) + kHi]);
        #pragma unroll
        for (int nt = 0; nt < 4; ++nt) {
            v2f b = *reinterpret_cast<const v2f*>(
                bbase + ((((k << 2) + nt) << 5) + lane) * 2);
            acc[nt] = __builtin_amdgcn_wmma_f32_16x16x4_f32(
                false, a, false, b, (short)0, acc[nt], false, false);
        }
    }

    if (active) {
        #pragma unroll
        for (int nt = 0; nt < 4; ++nt) {
            int n = (nt << 4) + (lane & 15);
            #pragma unroll
            for (int v = 0; v < 8; ++v) {
                int m = v + ((lane >> 4) << 3);
                if (m < cnt) {
                    float val = acc[nt][v] * normlds[w][m];
                    atomicAdd(&agg[(size_t)dstlds[w][m] * OUT_DIM + n], val);
                }
            }
        }
    }
}

// K8: ReLU into d_out
__global__ void rgcn_relu_kernel(const float* agg, float* out, int total) {
    int i = blockIdx.x * blockDim.x + threadIdx.x;
    if (i < total) out[i] = fmaxf(agg[i], 0.0f);
}

// ---------------------------------------------------------------------------
extern "C" void kernel_launch(void* const* d_in, const int* in_sizes, int n_in,
                              void* d_out, int out_size, void* d_ws, size_t ws_size,
                              hipStream_t stream) {
    const float* node_feat = (const float*)d_in[0];
    const float* weight    = (const float*)d_in[1];   // [8,64,64]
    const float* w_comp    = (const float*)d_in[2];   // [64,8]
    const float* loopw     = (const float*)d_in[3];   // [64,64]
    const float* bias      = (const float*)d_in[4];   // [64]
    const int*   src       = (const int*)d_in[5];
    const int*   dst       = (const int*)d_in[6];
    const int*   etype     = (const int*)d_in[7];

    const int N = in_sizes[0] / IN_DIM;
    const int E = in_sizes[5];

    // workspace layout (all 4-byte elements; relpack offset is 8B aligned)
    float* deg       = (float*)d_ws;                 // N floats (becomes norm)
    int*   counts    = (int*)(deg + N);              // 64
    int*   bin_start = counts + NUM_RELS;            // 65
    int*   tile_off  = bin_start + NUM_RELS + 1;     // 65
    int*   cursor    = tile_off + NUM_RELS + 1;      // 64
    int*   sorted    = cursor + NUM_RELS;            // E
    float* relpack   = (float*)(sorted + E);         // 64*4096
    float* looppack  = relpack + (size_t)NUM_RELS * 4096;  // 4096
    float* agg       = looppack + 4096;              // N*64

    const int T = 256;
    int zmax = (N > NUM_RELS) ? N : NUM_RELS;

    rgcn_zero_kernel<<<(zmax + T - 1) / T, T, 0, stream>>>(deg, counts, N);
    rgcn_deg_hist_kernel<<<(E + T - 1) / T, T, 0, stream>>>(dst, etype, deg, counts, E);
    rgcn_norm_kernel<<<(N + T - 1) / T, T, 0, stream>>>(deg, N);
    rgcn_scan_kernel<<<1, 32, 0, stream>>>(counts, bin_start, tile_off, cursor);
    rgcn_scatter_kernel<<<(E + T - 1) / T, T, 0, stream>>>(etype, cursor, sorted, E);
    rgcn_pack_kernel<<<NUM_RELS + 1, T, 0, stream>>>(weight, w_comp, loopw, relpack, looppack);

    int nodeTiles = (N + 15) >> 4;
    rgcn_selfloop_kernel<<<(nodeTiles + WAVES - 1) / WAVES, WAVES * 32, 0, stream>>>(
        node_feat, deg, looppack, bias, agg, N);

    int maxEdgeTiles = ((E + 15) >> 4) + NUM_RELS;   // upper bound on device tile count
    rgcn_edge_kernel<<<(maxEdgeTiles + WAVES - 1) / WAVES, WAVES * 32, 0, stream>>>(
        node_feat, src, dst, deg, sorted, bin_start, tile_off, relpack, agg);

    rgcn_relu_kernel<<<((size_t)N * OUT_DIM + T - 1) / T, T, 0, stream>>>(
        agg, (float*)d_out, N * OUT_DIM);
}